// SimpleGCN_68788196212904
// MI455X (gfx1250) — compile-verified
//
#include <hip/hip_runtime.h>

// ---------------------------------------------------------------------------
// SimpleGCN forward for MI455X (gfx1250, wave32).
// Dense GEMMs use V_WMMA_F32_16X16X4_F32 (fp32 matrix pipe, exact-precision
// path). Edge message passing is L2-resident scatter-add with f32 global
// atomics (unsafeAtomicAdd -> global_atomic_add_f32).
// ---------------------------------------------------------------------------

typedef float v2f __attribute__((ext_vector_type(2)));
typedef float v8f __attribute__((ext_vector_type(8)));

#define NN  100000   // nodes
#define NE  1600000  // edges
#define NG  256      // graphs
#define DH  64       // D_IN == D_H
#define DFC 128
#define NCLS 4

// ---------------- utility -------------------------------------------------

__global__ void fill_f32(float* __restrict__ p, float v, int n) {
  int i = blockIdx.x * blockDim.x + threadIdx.x;
  if (i < n) p[i] = v;
}

// deg[d] += 1 per edge (self-loop "1" pre-filled by fill_f32)
__global__ void deg_edges(const int* __restrict__ dst, float* __restrict__ deg, int nE) {
  int e = blockIdx.x * blockDim.x + threadIdx.x;
  if (e < nE) unsafeAtomicAdd(&deg[dst[e]], 1.0f);
}

__global__ void deg_to_dinv(float* __restrict__ deg, int n) {
  int i = blockIdx.x * blockDim.x + threadIdx.x;
  if (i < n) deg[i] = rsqrtf(deg[i]);   // deg >= 1 always (self loop)
}

// ---------------- WMMA GEMM: C[M,N] = A[M,K] @ B[K,N] (+bias)(+relu) ------
// M % 16 == 0, N % 16 == 0, K % 4 == 0. One wave per 16x16 output tile.
// ISA layouts (CDNA5 ISA §7.12.2):
//   A 16x4 f32 : lane = half*16+r holds A[r][k0+2*half], A[r][k0+2*half+1]
//   B 4x16 f32 : lane holds B[k0+2*half][c], B[k0+2*half+1][c]
//   C/D 16x16  : VGPR j, lane -> (M = j + 8*half, N = r)

__global__ void wmma_gemm_f32(const float* __restrict__ A,
                              const float* __restrict__ B,
                              const float* __restrict__ bias,
                              float* __restrict__ C,
                              int M, int K, int N, int fuse_relu)
{
  int tilesN  = N >> 4;
  int nTiles  = (M >> 4) * tilesN;
  int wave    = blockIdx.x * (blockDim.x >> 5) + (threadIdx.x >> 5);
  if (wave >= nTiles) return;            // wave-uniform: EXEC stays all-1s

  int tm   = wave / tilesN;
  int tn   = wave - tm * tilesN;
  int lane = threadIdx.x & 31;
  int half = lane >> 4;                  // 0 or 1
  int r    = lane & 15;

  const float* Arow = A + (size_t)(tm * 16 + r) * K;
  int col = tn * 16 + r;

  v8f acc = {};
  for (int k0 = 0; k0 < K; k0 += 4) {
    int kk = k0 + 2 * half;
    v2f a, b;
    a.x = Arow[kk];
    a.y = Arow[kk + 1];
    b.x = B[(size_t)kk * N + col];
    b.y = B[(size_t)(kk + 1) * N + col];
    acc = __builtin_amdgcn_wmma_f32_16x16x4_f32(
        /*neg_a=*/false, a, /*neg_b=*/false, b,
        /*c_mod=*/(short)0, acc, /*reuse_a=*/false, /*reuse_b=*/false);
  }

  float bval = bias ? bias[col] : 0.0f;
  #pragma unroll
  for (int j = 0; j < 8; ++j) {
    int row = tm * 16 + j + 8 * half;
    float v = acc[j] + bval;
    if (fuse_relu) v = fmaxf(v, 0.0f);
    C[(size_t)row * N + col] = v;
  }
}

// ---------------- edge scatter: agg[dst] += xw[src] * dinv[s]*dinv[d] -----
// One wave per edge, 2 columns per lane -> coalesced 256B row accesses.

__global__ void edge_scatter(const float* __restrict__ xw,
                             const int* __restrict__ src,
                             const int* __restrict__ dst,
                             const float* __restrict__ dinv,
                             float* __restrict__ agg, int nE)
{
  int t    = blockIdx.x * blockDim.x + threadIdx.x;
  int e    = t >> 5;
  int lane = t & 31;
  if (e >= nE) return;
  int s = src[e], d = dst[e];
  float nrm = dinv[s] * dinv[d];
  const float2 v = *(const float2*)(xw + (size_t)s * DH + lane * 2);
  float* out = agg + (size_t)d * DH + lane * 2;
  unsafeAtomicAdd(out,     v.x * nrm);
  unsafeAtomicAdd(out + 1, v.y * nrm);
}

// ---------------- finalize: h = relu(agg + dinv^2 * xw + b) (in place) ----

__global__ void gcn_finalize(float* __restrict__ agg,       // in/out
                             const float* __restrict__ xw,
                             const float* __restrict__ dinv,
                             const float* __restrict__ b, int nElem)
{
  int idx = blockIdx.x * blockDim.x + threadIdx.x;
  if (idx >= nElem) return;
  int i = idx >> 6;          // node
  int c = idx & 63;          // column
  float di = dinv[i];
  float v = agg[idx] + di * di * xw[idx] + b[c];
  agg[idx] = fmaxf(v, 0.0f);
}

// ---------------- global mean pool ----------------------------------------

__global__ void pool_accum(const float* __restrict__ h,
                           const int* __restrict__ batch,
                           float* __restrict__ gsum, int nElem)
{
  int idx = blockIdx.x * blockDim.x + threadIdx.x;
  if (idx >= nElem) return;
  int i = idx >> 6;
  int c = idx & 63;
  unsafeAtomicAdd(&gsum[(size_t)batch[i] * DH + c], h[idx]);
}

__global__ void pool_count(const int* __restrict__ batch, float* __restrict__ cnt, int n) {
  int i = blockIdx.x * blockDim.x + threadIdx.x;
  if (i < n) unsafeAtomicAdd(&cnt[batch[i]], 1.0f);
}

__global__ void pool_div(const float* __restrict__ gsum,
                         const float* __restrict__ cnt,
                         float* __restrict__ g, int nElem)
{
  int idx = blockIdx.x * blockDim.x + threadIdx.x;
  if (idx >= nElem) return;
  g[idx] = gsum[idx] / fmaxf(cnt[idx >> 6], 1.0f);
}

// ---------------- classifier: [NG,DFC] @ [DFC,NCLS] + bc ------------------

__global__ void head_classifier(const float* __restrict__ f2,
                                const float* __restrict__ Wc,
                                const float* __restrict__ bc,
                                float* __restrict__ out)
{
  int t = blockIdx.x * blockDim.x + threadIdx.x;
  if (t >= NG * NCLS) return;
  int g = t / NCLS, c = t - (t / NCLS) * NCLS;
  float s = bc[c];
  const float* row = f2 + (size_t)g * DFC;
  #pragma unroll 4
  for (int k = 0; k < DFC; ++k) s += row[k] * Wc[k * NCLS + c];
  out[t] = s;
}

// ---------------------------------------------------------------------------

static inline int cdiv(long long a, long long b) { return (int)((a + b - 1) / b); }

extern "C" void kernel_launch(void* const* d_in, const int* in_sizes, int n_in,
                              void* d_out, int out_size, void* d_ws, size_t ws_size,
                              hipStream_t stream) {
  // inputs (setup_inputs order)
  const float* x   = (const float*)d_in[0];
  const int*   ei  = (const int*)  d_in[1];   // [2, NE]
  const int*   bat = (const int*)  d_in[2];
  const float* W1  = (const float*)d_in[3];
  const float* b1  = (const float*)d_in[4];
  const float* W2  = (const float*)d_in[5];
  const float* b2  = (const float*)d_in[6];
  const float* Wf1 = (const float*)d_in[7];
  const float* bf1 = (const float*)d_in[8];
  const float* Wf2 = (const float*)d_in[9];
  const float* bf2 = (const float*)d_in[10];
  const float* Wc  = (const float*)d_in[11];
  const float* bc  = (const float*)d_in[12];
  float* out = (float*)d_out;

  const int* src = ei;
  const int* dst = ei + NE;

  // workspace carve-up (floats)
  float* buf0  = (float*)d_ws;             // [NN, 64]  xw
  float* buf1  = buf0 + (size_t)NN * DH;   // [NN, 64]  agg / h (in place)
  float* dinv  = buf1 + (size_t)NN * DH;   // [NN]      deg -> dinv
  float* gsum  = dinv + NN;                // [NG, 64]
  float* cnt   = gsum + NG * DH;           // [NG]      (contiguous with gsum)
  float* gpool = cnt  + NG;                // [NG, 64]
  float* f1    = gpool + NG * DH;          // [NG, 128]
  float* f2    = f1   + NG * DFC;          // [NG, 128]

  const int B = 256;
  const int elemsH = NN * DH;              // 6.4M

  // --- degree / normalization ---
  fill_f32   <<<cdiv(NN, B), B, 0, stream>>>(dinv, 1.0f, NN);     // self loops
  deg_edges  <<<cdiv(NE, B), B, 0, stream>>>(dst, dinv, NE);
  deg_to_dinv<<<cdiv(NN, B), B, 0, stream>>>(dinv, NN);

  // --- layer 1: xw = x @ W1 ; scatter ; finalize ---
  {
    int tiles = (NN / 16) * (DH / 16);
    wmma_gemm_f32<<<cdiv(tiles, 8), B, 0, stream>>>(x, W1, nullptr, buf0, NN, DH, DH, 0);
  }
  fill_f32    <<<cdiv(elemsH, B), B, 0, stream>>>(buf1, 0.0f, elemsH);
  edge_scatter<<<cdiv((long long)NE * 32, B), B, 0, stream>>>(buf0, src, dst, dinv, buf1, NE);
  gcn_finalize<<<cdiv(elemsH, B), B, 0, stream>>>(buf1, buf0, dinv, b1, elemsH);

  // --- layer 2: xw = h1 @ W2 ; scatter ; finalize ---
  {
    int tiles = (NN / 16) * (DH / 16);
    wmma_gemm_f32<<<cdiv(tiles, 8), B, 0, stream>>>(buf1, W2, nullptr, buf0, NN, DH, DH, 0);
  }
  fill_f32    <<<cdiv(elemsH, B), B, 0, stream>>>(buf1, 0.0f, elemsH);
  edge_scatter<<<cdiv((long long)NE * 32, B), B, 0, stream>>>(buf0, src, dst, dinv, buf1, NE);
  gcn_finalize<<<cdiv(elemsH, B), B, 0, stream>>>(buf1, buf0, dinv, b2, elemsH);

  // --- global mean pool ---
  fill_f32  <<<cdiv(NG * DH + NG, B), B, 0, stream>>>(gsum, 0.0f, NG * DH + NG); // gsum + cnt
  pool_accum<<<cdiv(elemsH, B), B, 0, stream>>>(buf1, bat, gsum, elemsH);
  pool_count<<<cdiv(NN, B), B, 0, stream>>>(bat, cnt, NN);
  pool_div  <<<cdiv(NG * DH, B), B, 0, stream>>>(gsum, cnt, gpool, NG * DH);

  // --- MLP head (WMMA with fused bias+relu) ---
  {
    int tiles = (NG / 16) * (DFC / 16);
    wmma_gemm_f32<<<cdiv(tiles, 8), B, 0, stream>>>(gpool, Wf1, bf1, f1, NG, DH, DFC, 1);
    wmma_gemm_f32<<<cdiv(tiles, 8), B, 0, stream>>>(f1, Wf2, bf2, f2, NG, DFC, DFC, 1);
  }
  head_classifier<<<cdiv(NG * NCLS, B), B, 0, stream>>>(f2, Wc, bc, out);
}